// CDFLossIndexPytorch_84241488544129
// MI455X (gfx1250) — compile-verified
//
#include <hip/hip_runtime.h>
#include <hip/hip_bf16.h>

#define NUM_BINS  512
#define NUM_NODES 128
#define KDIM      16384
#define NPAIRS    1024

typedef float v2f __attribute__((ext_vector_type(2)));
typedef float v8f __attribute__((ext_vector_type(8)));

// ---------------------------------------------------------------- zero ws
__global__ void zero_ws_kernel(float* __restrict__ p, int n) {
    int i = blockIdx.x * blockDim.x + threadIdx.x;
    if (i < n) p[i] = 0.0f;
}

// ---------------------------------------------------------------- histogram
// One block processes CHUNK contiguous elements of one pair-row.
// LDS-privatized 512-bin f32 histogram (ds_add_f32), flushed with global
// atomic_add_f32 into BOTH the src-node and dst-node rows.
__global__ __launch_bounds__(256) void hist_kernel(
    const float* __restrict__ residuals, const float* __restrict__ weights,
    const int* __restrict__ src, const int* __restrict__ dst,
    float* __restrict__ hist, float* __restrict__ tw,
    int chunks_per_row, int chunk) {
    __shared__ float s_hist[NUM_BINS];
    __shared__ float s_w;
    const int row = blockIdx.x / chunks_per_row;
    const int c   = blockIdx.x - row * chunks_per_row;
    const int t   = threadIdx.x;

    for (int b = t; b < NUM_BINS; b += 256) s_hist[b] = 0.0f;
    if (t == 0) s_w = 0.0f;
    __syncthreads();

    const float* __restrict__ rp = residuals + (size_t)row * KDIM + (size_t)c * chunk;
    const float* __restrict__ wp = weights   + (size_t)row * KDIM + (size_t)c * chunk;

    float wsum = 0.0f;
    for (int i = t; i < chunk; i += 256) {
        float rv = rp[i];
        float wv = wp[i];
        wsum += wv;                                   // tw sums ALL weights
        int bi = (int)floorf(rv * (float)NUM_BINS);   // BIN_W = 1/512 exactly
        if (bi >= 0 && bi < NUM_BINS) atomicAdd(&s_hist[bi], wv);
    }
    atomicAdd(&s_w, wsum);
    __syncthreads();

    const int s = src[row], d = dst[row];
    for (int b = t; b < NUM_BINS; b += 256) {
        float v = s_hist[b];
        atomicAdd(&hist[(size_t)s * NUM_BINS + b], v);
        atomicAdd(&hist[(size_t)d * NUM_BINS + b], v);
    }
    if (t == 0) {
        atomicAdd(&tw[s], s_w);
        atomicAdd(&tw[d], s_w);
    }
}

// ---------------------------------------------------------------- pmf+cumsum via WMMA
// cdf[n, 16J+j] = carry[n] + sum_{i<=j} pmf[n, 16J+i]
//
// Per 16-column block (one wave per 16-node tile):
//   D  = (A x Tri)  + CC    -> cdf tile (Tri[i][j] = i<=j)
//   CC = (A x Ones) + CC    -> every column of Ones-B yields the full row sum,
//                              so CC is the per-row carry already broadcast
//                              across all 16 columns: exactly the next C operand.
// No cross-lane shuffles needed; pure WMMA-to-WMMA dataflow.
__global__ __launch_bounds__(32) void cdf_kernel(
    const float* __restrict__ hist, const float* __restrict__ tw,
    float* __restrict__ cdf) {
    const int lane = threadIdx.x;        // 0..31
    const int grp  = lane >> 4;          // 0 or 1
    const int ln   = lane & 15;
    const int nodeBase = blockIdx.x * 16;

    // pmf = hist / (tw + 1e-10): fold the divide into a per-row scale on A.
    const float t_ = tw[nodeBase + ln];
    const float invtw = 1.0f / (t_ + 1e-10f);
    // A layout (16x4 f32): lanes 0-15 & 16-31 both hold row M=ln;
    // vgpr0/1 hold K = {0,1} (grp 0) or {2,3} (grp 1) within each 4-chunk.
    const float* __restrict__ histRow =
        hist + (size_t)(nodeBase + ln) * NUM_BINS + 2 * grp;

    // Triangular B slices for the 4 K-chunks, and the all-ones B.
    v2f btri[4];
#pragma unroll
    for (int kk = 0; kk < 4; ++kk) {
        const int i0 = 4 * kk + 2 * grp;
        btri[kk].x = (i0     <= ln) ? 1.0f : 0.0f;
        btri[kk].y = (i0 + 1 <= ln) ? 1.0f : 0.0f;
    }
    v2f bones;
    bones.x = 1.0f;
    bones.y = 1.0f;

    v8f cc = {};                                       // running carry accumulator

    for (int J = 0; J < NUM_BINS / 16; ++J) {
        v8f d   = cc;                                  // cdf tile accumulator
        v8f cc2 = cc;                                  // next-carry accumulator
#pragma unroll
        for (int kk = 0; kk < 4; ++kk) {
            const int col = 16 * J + 4 * kk;
            v2f a;
            a.x = histRow[col]     * invtw;
            a.y = histRow[col + 1] * invtw;
            d   = __builtin_amdgcn_wmma_f32_16x16x4_f32(
                      false, a, false, btri[kk], (short)0, d,   false, false);
            cc2 = __builtin_amdgcn_wmma_f32_16x16x4_f32(
                      false, a, false, bones,    (short)0, cc2, false, false);
        }
        cc = cc2;

        // store D: lane group grp, vgpr r -> row M = r + 8*grp, col N = ln
#pragma unroll
        for (int r = 0; r < 8; ++r) {
            const int m = r + 8 * grp;
            cdf[(size_t)(nodeBase + m) * NUM_BINS + 16 * J + ln] = d[r];
        }
    }
}

// ---------------------------------------------------------------- lookup (gather)
// float4-vectorized streaming pass; cdf table (256KB) stays hot in L2/WGP$.
__global__ __launch_bounds__(256) void lookup_kernel(
    const float* __restrict__ residuals, const float* __restrict__ weights,
    const int* __restrict__ src, const int* __restrict__ dst,
    const float* __restrict__ cdf,
    float* __restrict__ out_src, float* __restrict__ out_dst) {
    const size_t base = ((size_t)blockIdx.x * 256 + threadIdx.x) * 4;
    const int row = (int)(base >> 14);                // KDIM = 2^14
    const int s = src[row], d = dst[row];
    const float* __restrict__ cs = cdf + (size_t)s * NUM_BINS;
    const float* __restrict__ cd = cdf + (size_t)d * NUM_BINS;

    const float4 r4 = *(const float4*)(residuals + base);
    const float4 w4 = *(const float4*)(weights + base);
    float rr[4] = {r4.x, r4.y, r4.z, r4.w};
    float ww[4] = {w4.x, w4.y, w4.z, w4.w};
    float os[4], od[4];
#pragma unroll
    for (int i = 0; i < 4; ++i) {
        int bi = (int)floorf(rr[i] * (float)NUM_BINS + 0.5f);
        bool valid = (bi >= 0) & (bi < NUM_BINS) & (ww[i] > 0.0f);
        bi = bi < 0 ? 0 : (bi > NUM_BINS - 1 ? NUM_BINS - 1 : bi);
        float sv = cs[bi];
        float dv = cd[bi];
        os[i] = valid ? sv : 2.0f;
        od[i] = valid ? dv : 2.0f;
    }
    *(float4*)(out_src + base) = make_float4(os[0], os[1], os[2], os[3]);
    *(float4*)(out_dst + base) = make_float4(od[0], od[1], od[2], od[3]);
}

// ---------------------------------------------------------------- launch
extern "C" void kernel_launch(void* const* d_in, const int* in_sizes, int n_in,
                              void* d_out, int out_size, void* d_ws, size_t ws_size,
                              hipStream_t stream) {
    const float* residuals = (const float*)d_in[0];
    const float* weights   = (const float*)d_in[1];
    const int*   src       = (const int*)d_in[2];
    const int*   dst       = (const int*)d_in[3];

    float* hist = (float*)d_ws;                         // 128*512 floats
    float* tw   = hist + (size_t)NUM_NODES * NUM_BINS;  // 128 floats (pad to 256)
    float* cdf  = tw + 256;                             // 128*512 floats

    float* out_src = (float*)d_out;
    float* out_dst = out_src + (size_t)NPAIRS * KDIM;

    // zero hist + tw every call (harness does not re-poison between replays)
    const int nzero = NUM_NODES * NUM_BINS + 256;
    zero_ws_kernel<<<(nzero + 255) / 256, 256, 0, stream>>>(hist, nzero);

    const int CHUNK = 8192;                              // 32 elems/thread
    const int chunksPerRow = KDIM / CHUNK;               // 2
    hist_kernel<<<NPAIRS * chunksPerRow, 256, 0, stream>>>(
        residuals, weights, src, dst, hist, tw, chunksPerRow, CHUNK);

    cdf_kernel<<<NUM_NODES / 16, 32, 0, stream>>>(hist, tw, cdf);

    lookup_kernel<<<(NPAIRS * (size_t)KDIM) / 1024, 256, 0, stream>>>(
        residuals, weights, src, dst, cdf, out_src, out_dst);
}